// GCN_19559281066058
// MI455X (gfx1250) — compile-verified
//
#include <hip/hip_runtime.h>

typedef __attribute__((ext_vector_type(2))) float v2f;
typedef __attribute__((ext_vector_type(8))) float v8f;

#define EDIM 64

// ---------------------------------------------------------------------------
// Degree: deg[i] = 1 (self loop) + #edges targeting i
// ---------------------------------------------------------------------------
__global__ void gcn_deg_init(float* __restrict__ deg, int n) {
    int i = blockIdx.x * blockDim.x + threadIdx.x;
    if (i < n) deg[i] = 1.0f;
}

__global__ void gcn_deg_accum(const long long* __restrict__ col,
                              float* __restrict__ deg, int n_edges) {
    int e = blockIdx.x * blockDim.x + threadIdx.x;
    if (e < n_edges) atomicAdd(&deg[(int)col[e]], 1.0f);
}

__global__ void gcn_deg_to_dinv(float* __restrict__ deg, int n) {
    int i = blockIdx.x * blockDim.x + threadIdx.x;
    if (i < n) {
        float d = deg[i];
        deg[i] = (d > 0.0f) ? rsqrtf(d) : 0.0f;   // in place: deg -> dinv
    }
}

// ---------------------------------------------------------------------------
// xW = emb_table[nodes] @ W^T  via V_WMMA_F32_16X16X4_F32
// One wave per 16-row tile; 4 N-tiles (64 cols) accumulated over K=64.
// A frag (16x4 f32): lane = M%16, K = {2*half, 2*half+1}  -> float2 from x row
// B frag (4x16 f32): lane = N%16, K = {2*half, 2*half+1}  -> float2 from W row
//   (B = W^T so B[k][n] = W[n][k]; row n of W is contiguous)
// D (16x16 f32): VGPR r holds M = r + 8*half, N = lane%16
// ---------------------------------------------------------------------------
__global__ void gcn_gemm_wmma(const float* __restrict__ emb,
                              const float* __restrict__ W,
                              const int*   __restrict__ nodes,
                              float* __restrict__ xW, int n_rows) {
    int wave = (int)((blockIdx.x * blockDim.x + threadIdx.x) >> 5);
    int lane = threadIdx.x & 31;
    int row0 = wave * 16;               // wave-uniform tile base
    if (row0 >= n_rows) return;         // uniform per wave: EXEC stays all-1s

    int half = lane >> 4;               // 0: K=0,1   1: K=2,3
    int l16  = lane & 15;

    int src = nodes[row0 + l16];        // gather row index for this lane's A frag
    const float* xrow = emb + (long)src * EDIM;

    v8f acc0 = {}, acc1 = {}, acc2 = {}, acc3 = {};

    #pragma unroll
    for (int kb = 0; kb < EDIM; kb += 4) {
        int k = kb + 2 * half;
        v2f a  = *(const v2f*)(xrow + k);
        v2f b0 = *(const v2f*)(W + (long)( 0 + l16) * EDIM + k);
        v2f b1 = *(const v2f*)(W + (long)(16 + l16) * EDIM + k);
        v2f b2 = *(const v2f*)(W + (long)(32 + l16) * EDIM + k);
        v2f b3 = *(const v2f*)(W + (long)(48 + l16) * EDIM + k);
        acc0 = __builtin_amdgcn_wmma_f32_16x16x4_f32(false, a, false, b0, (short)0, acc0, false, false);
        acc1 = __builtin_amdgcn_wmma_f32_16x16x4_f32(false, a, false, b1, (short)0, acc1, false, false);
        acc2 = __builtin_amdgcn_wmma_f32_16x16x4_f32(false, a, false, b2, (short)0, acc2, false, false);
        acc3 = __builtin_amdgcn_wmma_f32_16x16x4_f32(false, a, false, b3, (short)0, acc3, false, false);
    }

    #pragma unroll
    for (int r = 0; r < 8; ++r) {
        int m = row0 + r + 8 * half;
        float* orow = xW + (long)m * EDIM;
        orow[ 0 + l16] = acc0[r];
        orow[16 + l16] = acc1[r];
        orow[32 + l16] = acc2[r];
        orow[48 + l16] = acc3[r];
    }
}

// ---------------------------------------------------------------------------
// out[i,:] = dinv[i]^2 * xW[i,:]   (self-loop term; also clears poison)
// ---------------------------------------------------------------------------
__global__ void gcn_self_loop_init(const float* __restrict__ xW,
                                   const float* __restrict__ dinv,
                                   float* __restrict__ out, int n_elems) {
    int idx = blockIdx.x * blockDim.x + threadIdx.x;
    if (idx < n_elems) {
        float d = dinv[idx >> 6];
        out[idx] = d * d * xW[idx];
    }
}

// ---------------------------------------------------------------------------
// Edge scatter: one wave per edge; each lane handles 2 of 64 columns.
// out[t,:] += dinv[s]*dinv[t] * xW[s,:]   via global_atomic_add_f32
// ---------------------------------------------------------------------------
__global__ void gcn_edge_scatter(const long long* __restrict__ row,
                                 const long long* __restrict__ col,
                                 const float* __restrict__ xW,
                                 const float* __restrict__ dinv,
                                 float* __restrict__ out, int n_edges) {
    int wave = (int)((blockIdx.x * blockDim.x + threadIdx.x) >> 5);
    if (wave >= n_edges) return;
    int lane = threadIdx.x & 31;

    int s = (int)row[wave];
    int t = (int)col[wave];
    float nrm = dinv[s] * dinv[t];

    const float* xr = xW + (long)s * EDIM;
    float*     orow = out + (long)t * EDIM;

    v2f v = *(const v2f*)(xr + 2 * lane);
    atomicAdd(orow + 2 * lane,     nrm * v.x);
    atomicAdd(orow + 2 * lane + 1, nrm * v.y);
}

// ---------------------------------------------------------------------------
// out = relu(out + b)
// ---------------------------------------------------------------------------
__global__ void gcn_bias_relu(float* __restrict__ out,
                              const float* __restrict__ b, int n_elems) {
    int idx = blockIdx.x * blockDim.x + threadIdx.x;
    if (idx < n_elems) {
        float v = out[idx] + b[idx & (EDIM - 1)];
        out[idx] = v > 0.0f ? v : 0.0f;
    }
}

// ---------------------------------------------------------------------------
// Launch
// inputs: 0=nodes(i32,N) 1=edges(i64,[2,E]) 2=emb(f32,N*64) 3=W(f32,64*64) 4=b(f32,64)
// ---------------------------------------------------------------------------
extern "C" void kernel_launch(void* const* d_in, const int* in_sizes, int n_in,
                              void* d_out, int out_size, void* d_ws, size_t ws_size,
                              hipStream_t stream) {
    const int*       nodes = (const int*)d_in[0];
    const long long* edges = (const long long*)d_in[1];
    const float*     emb   = (const float*)d_in[2];
    const float*     W     = (const float*)d_in[3];
    const float*     b     = (const float*)d_in[4];
    float*           out   = (float*)d_out;

    const int n_nodes = in_sizes[0];
    const int n_edges = in_sizes[1] / 2;
    const long long* row = edges;             // edges[0] = sources
    const long long* col = edges + n_edges;   // edges[1] = targets

    // workspace layout
    char* ws = (char*)d_ws;
    float* xW   = (float*)ws;                                      // n_nodes*64 f32
    float* dinv = (float*)(ws + (size_t)n_nodes * EDIM * sizeof(float)); // n_nodes f32 (deg->dinv)

    const int n_elems = n_nodes * EDIM;
    const int BS = 256;

    // 1) deg = 1 (self loops)
    gcn_deg_init<<<(n_nodes + BS - 1) / BS, BS, 0, stream>>>(dinv, n_nodes);
    // 2) deg += edge target counts
    gcn_deg_accum<<<(n_edges + BS - 1) / BS, BS, 0, stream>>>(col, dinv, n_edges);
    // 3) xW = emb[nodes] @ W^T   (WMMA; 1 wave per 16-row tile, 4 waves/block)
    {
        int tiles  = (n_nodes + 15) / 16;
        int blocks = (tiles + 3) / 4;         // 128 threads = 4 waves per block
        gcn_gemm_wmma<<<blocks, 128, 0, stream>>>(emb, W, nodes, xW, n_nodes);
    }
    // 4) dinv = rsqrt(deg)
    gcn_deg_to_dinv<<<(n_nodes + BS - 1) / BS, BS, 0, stream>>>(dinv, n_nodes);
    // 5) out = dinv^2 * xW (self loop)
    gcn_self_loop_init<<<(n_elems + BS - 1) / BS, BS, 0, stream>>>(xW, dinv, out, n_elems);
    // 6) scatter-add edge messages (1 wave per edge)
    {
        long long threads = (long long)n_edges * 32;
        int blocks = (int)((threads + BS - 1) / BS);
        gcn_edge_scatter<<<blocks, BS, 0, stream>>>(row, col, xW, dinv, out, n_edges);
    }
    // 7) out = relu(out + b)
    gcn_bias_relu<<<(n_elems + BS - 1) / BS, BS, 0, stream>>>(out, b, n_elems);
}